// ConcatAttention_11081015623955
// MI455X (gfx1250) — compile-verified
//
#include <hip/hip_runtime.h>
#include <hip/hip_bf16.h>

// Problem constants (fixed by the reference).
#define BDIM   8
#define HDIM   16
#define BH     128      // B*H
#define LK     4096
#define DD     128
#define ROWS   128      // K-rows per block in main kernel
#define WSTR   132      // padded LDS stride (128 dwords + 4 pad dwords)

typedef float v2f __attribute__((ext_vector_type(2)));
typedef float v8f __attribute__((ext_vector_type(8)));

typedef unsigned int u32x4 __attribute__((ext_vector_type(4)));
typedef int          i32x8 __attribute__((ext_vector_type(8)));
typedef int          i32x4 __attribute__((ext_vector_type(4)));

// ---------------------------------------------------------------------------
// TDM: async 2-D f32 tile load Global -> LDS with row padding.
//   dim0/tile0 : row length (elements), dim1/tile1 : #rows
//   stride0    : row pitch in elements (global)
//   padIntvEnc : pad_interval encoding (6 -> pad after every 128 dwords)
//   padDwords  : pad amount in dwords (0 = no padding)
// Descriptor layout per CDNA5 ISA ch.8 (D# groups 0..3).
// This toolchain exposes the 6-arg builtin:
//   (uint32x4 g0, int32x8 g1, int32x4 g2, int32x4 g3, int32x8 g4, i32 cpol)
// ---------------------------------------------------------------------------
__device__ static inline void tdm_load_tile_f32(unsigned int lds_off,
                                                const float* gptr,
                                                unsigned int dim0,
                                                unsigned int dim1,
                                                unsigned int tile0,
                                                unsigned int tile1,
                                                unsigned int stride0,
                                                unsigned int padIntvEnc,
                                                unsigned int padDwords)
{
    unsigned long long ga = (unsigned long long)(uintptr_t)gptr;

    u32x4 g0;
    g0[0] = 1u;                                        // count=1 (valid user D#)
    g0[1] = lds_off;                                   // lds_addr
    g0[2] = (unsigned int)ga;                          // global_addr[31:0]
    g0[3] = (unsigned int)((ga >> 32) & 0x01FFFFFFu)   // global_addr[56:32]
          | (2u << 30);                                // type = 2 ("image")

    unsigned int w0 = (2u << 16);                      // data_size = 4 bytes
    if (padDwords)
        w0 |= (1u << 20)                               // pad_enable
            | (padIntvEnc << 22)                       // pad_interval
            | ((padDwords - 1u) << 25);                // pad_amount (0 => 1 dword)

    i32x8 g1;
    g1[0] = (int)w0;
    g1[1] = (int)(dim0 << 16);                         // tensor_dim0[15:0] @bit48
    g1[2] = (int)((dim0 >> 16) | (dim1 << 16));        // dim0 hi | dim1 lo
    g1[3] = (int)((dim1 >> 16) | (tile0 << 16));       // dim1 hi | tile_dim0
    g1[4] = (int)(tile1 & 0xFFFFu);                    // tile_dim1 (tile_dim2=0)
    g1[5] = (int)stride0;                              // tensor_dim0_stride[31:0]
    g1[6] = 0;                                         // stride0 hi | stride1 lo
    g1[7] = 0;                                         // stride1 hi

    i32x4 z4 = {0, 0, 0, 0};                           // 2-D: groups 2/3 unused
    i32x8 z8 = {0, 0, 0, 0, 0, 0, 0, 0};
    __builtin_amdgcn_tensor_load_to_lds(g0, g1, z4, z4, z8, 0);
}

// ---------------------------------------------------------------------------
// Kernel A: tq[bh, :] = Q[bh, :] @ W_q     (M=128, K=128, N=128)
// One block, 256 threads = 8 waves; wave w owns rows [16w, 16w+16).
// ---------------------------------------------------------------------------
__global__ void __launch_bounds__(256)
tq_kernel(const float* __restrict__ Q, const float* __restrict__ Wq,
          float* __restrict__ tqOut)
{
    const int t    = threadIdx.x;
    const int lane = t & 31;
    const int w    = t >> 5;
    const int l16  = lane & 15;
    const int hi   = lane >> 4;

    const int rowBase = w * 16;
    const float* Qrow = Q + (size_t)rowBase * DD;

    v8f acc[8] = {};

    for (int kk = 0; kk < 32; ++kk) {
        const int k0 = kk * 4;
        const float* ap = Qrow + (size_t)l16 * DD + k0 + 2 * hi;
        v2f a;
        a.x = ap[0];
        a.y = ap[1];
#pragma unroll
        for (int nt = 0; nt < 8; ++nt) {
            const int brow = k0 + 2 * hi;
            v2f b;
            b.x = Wq[(size_t)(brow + 0) * DD + nt * 16 + l16];
            b.y = Wq[(size_t)(brow + 1) * DD + nt * 16 + l16];
            acc[nt] = __builtin_amdgcn_wmma_f32_16x16x4_f32(
                false, a, false, b, (short)0, acc[nt], false, false);
        }
    }

#pragma unroll
    for (int nt = 0; nt < 8; ++nt)
#pragma unroll
        for (int r = 0; r < 8; ++r)
            tqOut[(size_t)(rowBase + r + 8 * hi) * DD + nt * 16 + l16] = acc[nt][r];
}

// ---------------------------------------------------------------------------
// Kernel B: per (bh, 128-row K tile):
//   pre  = K @ W_pre + b_pre                        -> output
//   energy[k] = dot(tanh(pre[k,:] + tq[bh,:]), W_v) -> workspace
// W_pre and the K tile are DMA'd into padded LDS by the Tensor Data Mover.
// ---------------------------------------------------------------------------
__global__ void __launch_bounds__(256)
pre_energy_kernel(const float* __restrict__ K, const float* __restrict__ Wpre,
                  const float* __restrict__ bpre, const float* __restrict__ Wv,
                  const float* __restrict__ tq, float* __restrict__ preOut,
                  float* __restrict__ energy)
{
    __shared__ float sW[DD * WSTR];   // 67,584 B : padded W_pre
    __shared__ float sK[DD * WSTR];   // 67,584 B : padded K tile
    __shared__ float sB[DD];          // b_pre
    __shared__ float sT[DD];          // tq row for this bh
    __shared__ float sV[DD];          // W_v

    const int bh = blockIdx.y;
    const int rt = blockIdx.x;
    const int t  = threadIdx.x;

    // Wave 0 issues both TDM tile loads; pad_interval=6 (every 128 dwords),
    // pad 4 dwords -> LDS row pitch of WSTR=132 floats.
    if (t < 32) {
        const float* ktile = K + ((size_t)bh * LK + (size_t)rt * ROWS) * DD;
        tdm_load_tile_f32((unsigned int)(uintptr_t)&sW[0], Wpre,
                          DD, DD, DD, DD, DD, 6u, 4u);
        tdm_load_tile_f32((unsigned int)(uintptr_t)&sK[0], ktile,
                          DD, DD, DD, DD, DD, 6u, 4u);
    }
    if (t < DD) {
        sB[t] = bpre[t];
        sT[t] = tq[(size_t)bh * DD + t];
        sV[t] = Wv[t];
    }
    if (t < 32)
        __builtin_amdgcn_s_wait_tensorcnt(0);
    __syncthreads();

    const int lane = t & 31;
    const int w    = t >> 5;
    const int l16  = lane & 15;
    const int hi   = lane >> 4;

    const int rowBase = rt * ROWS + w * 16;

    // Accumulators initialized with bias (bias depends on column only).
    v8f acc[8];
#pragma unroll
    for (int nt = 0; nt < 8; ++nt) {
        const float bv = sB[nt * 16 + l16];
#pragma unroll
        for (int r = 0; r < 8; ++r) acc[nt][r] = bv;
    }

    const float* kbase = &sK[(w * 16 + l16) * WSTR];

    for (int kk = 0; kk < 32; ++kk) {
        const int k0 = kk * 4;
        // A fragment from LDS: lane (l16,hi) -> row (w*16+l16), cols k0+2hi+{0,1}
        // bank = (4*l16 + k0 + 2*hi) mod 64 -> 64 dwords across 64 banks.
        const float* ap = kbase + k0 + 2 * hi;
        v2f a;
        a.x = ap[0];
        a.y = ap[1];
        const int brow = k0 + 2 * hi;
#pragma unroll
        for (int nt = 0; nt < 8; ++nt) {
            v2f b;
            b.x = sW[(brow + 0) * WSTR + nt * 16 + l16];
            b.y = sW[(brow + 1) * WSTR + nt * 16 + l16];
            acc[nt] = __builtin_amdgcn_wmma_f32_16x16x4_f32(
                false, a, false, b, (short)0, acc[nt], false, false);
        }
    }

    // Store pre; fold tanh(pre + tq) * W_v into per-row partial sums.
    float* preBase = preOut + ((size_t)bh * LK + rowBase) * DD;
    float esum[8];
#pragma unroll
    for (int r = 0; r < 8; ++r) esum[r] = 0.0f;

#pragma unroll
    for (int nt = 0; nt < 8; ++nt) {
        const float tqv = sT[nt * 16 + l16];
        const float wvv = sV[nt * 16 + l16];
#pragma unroll
        for (int r = 0; r < 8; ++r) {
            const float p = acc[nt][r];
            preBase[(size_t)(r + 8 * hi) * DD + nt * 16 + l16] = p;
            esum[r] += tanhf(p + tqv) * wvv;
        }
    }

    // Reduce across the 16 lanes of each half (xor 8,4,2,1 stays in-half).
#pragma unroll
    for (int off = 8; off >= 1; off >>= 1)
#pragma unroll
        for (int r = 0; r < 8; ++r)
            esum[r] += __shfl_xor(esum[r], off, 32);

    if (l16 == 0) {
        float* eb = energy + (size_t)bh * LK + rowBase;
#pragma unroll
        for (int r = 0; r < 8; ++r) eb[r + 8 * hi] = esum[r];
    }
}

// ---------------------------------------------------------------------------
// Kernel C: per (bh): masked softmax over LK (score output) and
//           context = score @ V.
// ---------------------------------------------------------------------------
__global__ void __launch_bounds__(256)
softmax_context_kernel(const float* __restrict__ energy,
                       const int* __restrict__ mask,
                       const float* __restrict__ V,
                       float* __restrict__ ctxOut,
                       float* __restrict__ scoreOut)
{
    __shared__ float sScore[LK];      // 16 KB
    __shared__ float sRed[256];
    __shared__ float sCtx[256];

    const int bh = blockIdx.x;
    const int t  = threadIdx.x;

    const float* e = energy + (size_t)bh * LK;
    const int*   m = mask   + (size_t)bh * LK;

    float lmax = -1e30f;
    for (int k = t; k < LK; k += 256) {
        const float mv = (float)m[k];
        const float ev = e[k] * (1.0f - mv) + mv * (-1.0e6f);
        sScore[k] = ev;
        lmax = fmaxf(lmax, ev);
    }
    sRed[t] = lmax;
    __syncthreads();
    for (int s = 128; s > 0; s >>= 1) {
        if (t < s) sRed[t] = fmaxf(sRed[t], sRed[t + s]);
        __syncthreads();
    }
    const float gmax = sRed[0];
    __syncthreads();

    float lsum = 0.0f;
    for (int k = t; k < LK; k += 256) {
        const float x = expf(sScore[k] - gmax);
        sScore[k] = x;
        lsum += x;
    }
    sRed[t] = lsum;
    __syncthreads();
    for (int s = 128; s > 0; s >>= 1) {
        if (t < s) sRed[t] += sRed[t + s];
        __syncthreads();
    }
    const float inv = 1.0f / sRed[0];
    __syncthreads();

    float* so = scoreOut + (size_t)bh * LK;
    for (int k = t; k < LK; k += 256) {
        const float sc = sScore[k] * inv;
        sScore[k] = sc;
        so[k] = sc;
    }
    __syncthreads();

    // context[col] = sum_k score[k] * V[k][col]; k-range split over 2 halves.
    const float* Vb = V + (size_t)bh * LK * DD;
    const int col  = t & (DD - 1);
    const int half = t >> 7;
    float c = 0.0f;
    const int k0 = half * (LK / 2);
    for (int k = k0; k < k0 + LK / 2; ++k)
        c += sScore[k] * Vb[(size_t)k * DD + col];
    sCtx[t] = c;
    __syncthreads();
    if (t < DD)
        ctxOut[(size_t)bh * DD + t] = sCtx[t] + sCtx[t + DD];
}

// ---------------------------------------------------------------------------
extern "C" void kernel_launch(void* const* d_in, const int* in_sizes, int n_in,
                              void* d_out, int out_size, void* d_ws, size_t ws_size,
                              hipStream_t stream)
{
    (void)in_sizes; (void)n_in; (void)out_size; (void)ws_size;

    const float* Q    = (const float*)d_in[0];
    const float* K    = (const float*)d_in[1];
    const float* V    = (const float*)d_in[2];
    const int*   mask = (const int*)  d_in[3];
    // d_in[4] = scale (unused by reference forward)
    const float* Wpre = (const float*)d_in[5];
    const float* bpre = (const float*)d_in[6];
    const float* Wq   = (const float*)d_in[7];
    const float* Wv   = (const float*)d_in[8];

    // Output layout: context [BH*D], score [BH*LK], pre [BH*LK*D]
    float* ctxOut   = (float*)d_out;
    float* scoreOut = ctxOut + (size_t)BH * DD;
    float* preOut   = scoreOut + (size_t)BH * LK;

    // Workspace: tq [BH*D], energy [BH*LK]
    float* tqWs     = (float*)d_ws;
    float* energyWs = tqWs + (size_t)BH * DD;

    tq_kernel<<<1, 256, 0, stream>>>(Q, Wq, tqWs);

    dim3 grid(LK / ROWS, BH);
    pre_energy_kernel<<<grid, 256, 0, stream>>>(K, Wpre, bpre, Wv, tqWs,
                                                preOut, energyWs);

    softmax_context_kernel<<<BH, 256, 0, stream>>>(energyWs, mask, V,
                                                   ctxOut, scoreOut);
}